// TransformerSetAbstraction_33655363732079
// MI455X (gfx1250) — compile-verified
//
#include <hip/hip_runtime.h>
#include <hip/hip_bf16.h>
#include <stddef.h>

// ---------------------------------------------------------------------------
// Problem constants (from reference): B=16, N=4096, D=64, dm=128, Cin=67,
// S=NPOINT=1024, K=NSAMPLE=32, HEADS=4, dh=32, RADIUS=0.2
// ---------------------------------------------------------------------------
#define NB    16
#define NN    4096
#define DD    64
#define DM    128
#define CIN   67
#define CINP  68          // Cin padded to multiple of 4 for WMMA K-loop
#define SS    1024
#define KK    32
#define R2    0.04f
#define QSCL  0.1767766952966369f   // 1/sqrt(32)
#define STp   132         // LDS row stride for [32][128] buffers (132%64=4 -> no bank conflicts)
#define SCS   34          // scores row stride (even -> 8B-aligned v2f, conflict-free)

typedef __attribute__((ext_vector_type(2))) float v2f;
typedef __attribute__((ext_vector_type(8))) float v8f;

// D = A(16x4 f32) * B(4x16 f32) + C   — fp32 WMMA, matches reference precision.
__device__ __forceinline__ v8f wmma4(v2f a, v2f b, v8f c) {
  return __builtin_amdgcn_wmma_f32_16x16x4_f32(
      /*neg_a=*/false, a, /*neg_b=*/false, b,
      /*c_mod=*/(short)0, c, /*reuse_a=*/false, /*reuse_b=*/false);
}

__device__ __forceinline__ v2f ld2(const float* p) { return *(const v2f*)p; }

// ---------------------------------------------------------------------------
// Kernel 0: swizzle weights into WMMA B-fragment order in workspace:
//   Wsw[((t*2+p)*128 + n)*2 + c] = W[4t + 2p + c][n]
// so lane (p,j) of tile nb loads its fragment as ONE contiguous b64 at
// ((t*2+p)*128 + nb + j)*2.  Wq/Wk/Wv padded K 67->68; Wp1 padded 3->4 with
// row 3 = bp1 (bias folded via homogeneous coordinate).
// ---------------------------------------------------------------------------
__global__ __launch_bounds__(256) void prep_kernel(
    const float* __restrict__ Wq,  const float* __restrict__ Wk,
    const float* __restrict__ Wv,  const float* __restrict__ Wp1,
    const float* __restrict__ bp1, const float* __restrict__ Wp2,
    const float* __restrict__ Wo,
    float* __restrict__ WqS, float* __restrict__ WkS, float* __restrict__ WvS,
    float* __restrict__ Wp1S, float* __restrict__ Wp2S, float* __restrict__ WoS) {
  const int SZ68  = (CINP / 4) * 512;   // 17 k-steps * 512 = 8704
  const int SZ4   = 512;                //  1 k-step
  const int SZ128 = 32 * 512;           // 32 k-steps = 16384
  int o = blockIdx.x * 256 + threadIdx.x;

  int u, r, n;
  if (o < 3 * SZ68) {
    int m = o / SZ68; u = o % SZ68;
    r = (u >> 9) * 4 + ((u >> 8) & 1) * 2 + (u & 1); n = (u >> 1) & 127;
    const float* W = (m == 0) ? Wq : ((m == 1) ? Wk : Wv);
    float*       O = (m == 0) ? WqS : ((m == 1) ? WkS : WvS);
    O[u] = (r < CIN) ? W[r * DM + n] : 0.0f;
  } else if (o < 3 * SZ68 + SZ4) {
    u = o - 3 * SZ68;
    r = (u >> 9) * 4 + ((u >> 8) & 1) * 2 + (u & 1); n = (u >> 1) & 127;
    Wp1S[u] = (r < 3) ? Wp1[r * DM + n] : bp1[n];
  } else if (o < 3 * SZ68 + SZ4 + SZ128) {
    u = o - 3 * SZ68 - SZ4;
    r = (u >> 9) * 4 + ((u >> 8) & 1) * 2 + (u & 1); n = (u >> 1) & 127;
    Wp2S[u] = Wp2[r * DM + n];
  } else if (o < 3 * SZ68 + SZ4 + 2 * SZ128) {
    u = o - 3 * SZ68 - SZ4 - SZ128;
    r = (u >> 9) * 4 + ((u >> 8) & 1) * 2 + (u & 1); n = (u >> 1) & 127;
    WoS[u] = Wo[r * DM + n];
  }
}

// ---------------------------------------------------------------------------
// Kernel 1: farthest point sampling. One block per batch, 1024 threads,
// xyz staged in LDS (48KB), per-thread dist in registers, argmax via
// wave32 shuffle tree + cross-wave LDS reduce. Ties pick smallest index
// (matches jnp.argmax "first max").
// ---------------------------------------------------------------------------
__global__ __launch_bounds__(1024) void fps_kernel(
    const float* __restrict__ xyz, int* __restrict__ fps_idx) {
  __shared__ float xs[NN], ys[NN], zs[NN];
  __shared__ float redV[32];
  __shared__ int   redI[32];
  __shared__ int   bcast;

  const int b = blockIdx.x;
  const int tid = threadIdx.x;
  const int lane = tid & 31, wv = tid >> 5;
  const float* xb = xyz + (size_t)b * 3 * NN;

  for (int n = tid; n < NN; n += 1024) {
    xs[n] = xb[n];
    ys[n] = xb[NN + n];
    zs[n] = xb[2 * NN + n];
  }
  float dist[4];
#pragma unroll
  for (int k = 0; k < 4; ++k) dist[k] = 1e10f;
  __syncthreads();

  int far = 0;
  for (int i = 0; i < SS; ++i) {
    if (tid == 0) fps_idx[b * SS + i] = far;
    float cx = xs[far], cy = ys[far], cz = zs[far];
    float bestV = -1.0f; int bestI = 0;
#pragma unroll
    for (int k = 0; k < 4; ++k) {
      int n = tid + k * 1024;
      float dx = xs[n] - cx, dy = ys[n] - cy, dz = zs[n] - cz;
      float d = dx * dx + dy * dy + dz * dz;
      float dmn = fminf(dist[k], d);
      dist[k] = dmn;
      if (dmn > bestV || (dmn == bestV && n < bestI)) { bestV = dmn; bestI = n; }
    }
#pragma unroll
    for (int off = 16; off > 0; off >>= 1) {
      float v = __shfl_xor(bestV, off, 32);
      int   j = __shfl_xor(bestI, off, 32);
      if (v > bestV || (v == bestV && j < bestI)) { bestV = v; bestI = j; }
    }
    if (lane == 0) { redV[wv] = bestV; redI[wv] = bestI; }
    __syncthreads();
    if (wv == 0) {
      bestV = redV[lane]; bestI = redI[lane];
#pragma unroll
      for (int off = 16; off > 0; off >>= 1) {
        float v = __shfl_xor(bestV, off, 32);
        int   j = __shfl_xor(bestI, off, 32);
        if (v > bestV || (v == bestV && j < bestI)) { bestV = v; bestI = j; }
      }
      if (lane == 0) bcast = bestI;
    }
    __syncthreads();
    far = bcast;
  }
}

// ---------------------------------------------------------------------------
// Kernel 2: ball query (+ write new_xyz output). One wave per (b,s) query:
// scan N in index order, ordered compaction of first K in-ball indices via
// ballot + prefix popcount; pad tail with first in-ball index.
// ---------------------------------------------------------------------------
__global__ __launch_bounds__(256) void ballq_kernel(
    const float* __restrict__ xyz, const int* __restrict__ fps_idx,
    int* __restrict__ ball_idx, float* __restrict__ out_newxyz) {
  const int gw = blockIdx.x * 8 + (threadIdx.x >> 5);
  const int lane = threadIdx.x & 31;
  const int b = gw >> 10, s = gw & 1023;
  const float* xb = xyz + (size_t)b * 3 * NN;

  const int ci = fps_idx[b * SS + s];
  const float cx = xb[ci], cy = xb[NN + ci], cz = xb[2 * NN + ci];
  if (lane < 3) out_newxyz[((size_t)b * 3 + lane) * SS + s] = xb[lane * NN + ci];

  int* ob = ball_idx + (size_t)(b * SS + s) * KK;
  int count = 0, firstIdx = -1;
  for (int n0 = 0; n0 < NN && count < KK; n0 += 32) {
    int n = n0 + lane;
    float dx = xb[n] - cx, dy = xb[NN + n] - cy, dz = xb[2 * NN + n] - cz;
    bool inb = (dx * dx + dy * dy + dz * dz) <= R2;
    unsigned mm = (unsigned)__ballot(inb);           // wave32: low 32 bits
    if (firstIdx < 0 && mm) firstIdx = n0 + (__ffs(mm) - 1);
    int pos = count + __popc(mm & ((1u << lane) - 1u));
    if (inb && pos < KK) ob[pos] = n;
    count += __popc(mm);
  }
  if (count > KK) count = KK;
  if (lane >= count) ob[lane] = firstIdx;   // centroid guarantees count>=1
}

// ---------------------------------------------------------------------------
// Kernel 3: fused per-(b,s) transformer. 256 threads = 8 waves; wave w owns
// column tile nt=w of every 32x128 GEMM (both M-tiles, shared B fragment),
// and (head=w>>1, mtile=w&1) for the 32x32 attention GEMMs (shared A frag).
// All matrix math on v_wmma_f32_16x16x4_f32; every fragment but phase 6's V (two b32 loads) is one b64 load.
// ---------------------------------------------------------------------------
__global__ __launch_bounds__(256) void tsa_main(
    const float* __restrict__ xyz, const float* __restrict__ points,
    const float* __restrict__ WqS, const float* __restrict__ WkS,
    const float* __restrict__ WvS, const float* __restrict__ WoS,
    const float* __restrict__ Wp1S, const float* __restrict__ Wp2S,
    const float* __restrict__ bp2,
    const int* __restrict__ fps_idx, const int* __restrict__ ball_idx,
    float* __restrict__ out_pts) {
  extern __shared__ float sm[];
  float* bQ = sm;                    // [32][STp]  h1 -> q
  float* bK = bQ + 32 * STp;         // [32][STp]  k
  float* bV = bK + 32 * STp;         // [32][STp]  v
  float* bP = bV + 32 * STp;         // [32][STp]  pos -> attn@v output
  float* sc = bP + 32 * STp;         // [4*32][SCS] per-head scores/attn
  float* ft = sc + 4 * 32 * SCS;     // [32][68]   feat (zero-padded col 67)
  float* gx = ft + 32 * CINP;        // [32][4]    grouped_xyz (homog. col3=1)
  int*   id = (int*)(gx + 128);      // [32]       neighbor indices

  const int tid  = threadIdx.x;
  const int lane = tid & 31;
  const int w    = tid >> 5;
  const int p    = lane >> 4;        // fragment half-wave selector
  const int j    = lane & 15;        // fragment row/col within 16
  const int bs   = blockIdx.x;
  const int b    = bs >> 10;
  const int s    = bs & 1023;

  const float* xb = xyz + (size_t)b * 3 * NN;
  const float* pb = points + (size_t)b * DD * NN;

  if (tid < 32) id[tid] = ball_idx[(size_t)bs * KK + tid];
  __syncthreads();

  const int ci = fps_idx[bs];
  const float c0 = xb[ci], c1 = xb[NN + ci], c2 = xb[2 * NN + ci];

  // ---- gather: feat[j][0..2]=gxyz-center, [3..66]=points, [67]=0 ----------
  for (int e = tid; e < 32 * CINP; e += 256) {
    int jj = e & 31, c = e >> 5;
    int n = id[jj];
    float v;
    if (c < 3)        v = xb[c * NN + n] - (c == 0 ? c0 : (c == 1 ? c1 : c2));
    else if (c < CIN) v = pb[(c - 3) * NN + n];
    else              v = 0.0f;
    ft[jj * CINP + c] = v;
  }
  if (tid < 128) {
    int jj = tid & 31, c = tid >> 5;
    gx[jj * 4 + c] = (c < 3) ? xb[c * NN + id[jj]] : 1.0f;
  }
  __syncthreads();

  const int nb = w * 16;
  const int h = w >> 1, mt = w & 1, hb = h * 32;

  // ---- phase 1: h1 = relu(gxyz_h[32,4] @ Wp1p[4,128])  (bias folded) -----
  {
    v2f bf = ld2(Wp1S + (p * 128 + nb + j) * 2);
    v2f a0 = ld2(gx + j * 4 + 2 * p);
    v2f a1 = ld2(gx + (16 + j) * 4 + 2 * p);
    v8f d0 = {}, d1 = {};
    d0 = wmma4(a0, bf, d0);
    d1 = wmma4(a1, bf, d1);
#pragma unroll
    for (int r = 0; r < 8; ++r) {
      bQ[(r + 8 * p) * STp + nb + j]      = fmaxf(d0[r], 0.0f);
      bQ[(16 + r + 8 * p) * STp + nb + j] = fmaxf(d1[r], 0.0f);
    }
  }
  __syncthreads();

  // ---- phase 2: pos = h1[32,128] @ Wp2[128,128] + bp2 --------------------
  {
    v8f d0 = {}, d1 = {};
    for (int t = 0; t < 32; ++t) {
      v2f bb = ld2(Wp2S + ((t * 2 + p) * 128 + nb + j) * 2);
      v2f a0 = ld2(bQ + j * STp + 4 * t + 2 * p);
      v2f a1 = ld2(bQ + (16 + j) * STp + 4 * t + 2 * p);
      d0 = wmma4(a0, bb, d0);
      d1 = wmma4(a1, bb, d1);
    }
    float bias = bp2[nb + j];
#pragma unroll
    for (int r = 0; r < 8; ++r) {
      bP[(r + 8 * p) * STp + nb + j]      = d0[r] + bias;
      bP[(16 + r + 8 * p) * STp + nb + j] = d1[r] + bias;
    }
  }
  __syncthreads();

  // ---- phase 3: q = feat@Wq * 1/sqrt(dh);  k,v = feat@W + pos ------------
  {
    const float* Ws[3] = {WqS, WkS, WvS};
    float*       Ds[3] = {bQ, bK, bV};
    for (int m = 0; m < 3; ++m) {
      const float* Wm = Ws[m];
      float*       Dm = Ds[m];
      v8f d0 = {}, d1 = {};
      for (int t = 0; t < CINP / 4; ++t) {
        v2f bb = ld2(Wm + ((t * 2 + p) * 128 + nb + j) * 2);
        v2f a0 = ld2(ft + j * CINP + 4 * t + 2 * p);
        v2f a1 = ld2(ft + (16 + j) * CINP + 4 * t + 2 * p);
        d0 = wmma4(a0, bb, d0);
        d1 = wmma4(a1, bb, d1);
      }
#pragma unroll
      for (int r = 0; r < 8; ++r) {
        int r0 = (r + 8 * p) * STp + nb + j;
        int r1 = (16 + r + 8 * p) * STp + nb + j;
        if (m == 0) { Dm[r0] = d0[r] * QSCL;    Dm[r1] = d1[r] * QSCL; }
        else        { Dm[r0] = d0[r] + bP[r0];  Dm[r1] = d1[r] + bP[r1]; }
      }
    }
  }
  __syncthreads();

  // ---- phase 4: scores_h = q_h @ k_h^T  (A shared across both N-tiles) ---
  {
    v8f d0 = {}, d1 = {};
    for (int t = 0; t < 8; ++t) {
      v2f a  = ld2(bQ + (16 * mt + j) * STp + hb + 4 * t + 2 * p);
      v2f b0 = ld2(bK + j * STp + hb + 4 * t + 2 * p);          // k^T tiles
      v2f b1 = ld2(bK + (16 + j) * STp + hb + 4 * t + 2 * p);
      d0 = wmma4(a, b0, d0);
      d1 = wmma4(a, b1, d1);
    }
#pragma unroll
    for (int r = 0; r < 8; ++r) {
      int row = (h * 32 + 16 * mt + r + 8 * p) * SCS;
      sc[row + j]      = d0[r];
      sc[row + 16 + j] = d1[r];
    }
  }
  __syncthreads();

  // ---- phase 5: row softmax over the 32 neighbors (per head) -------------
  if (tid < 128) {
    int hh = tid >> 5, row = tid & 31;
    float* r0 = sc + (hh * 32 + row) * SCS;
    float mx = r0[0];
    for (int i = 1; i < 32; ++i) mx = fmaxf(mx, r0[i]);
    float sum = 0.0f;
    for (int i = 0; i < 32; ++i) { float e = expf(r0[i] - mx); r0[i] = e; sum += e; }
    float inv = 1.0f / sum;
    for (int i = 0; i < 32; ++i) r0[i] *= inv;
  }
  __syncthreads();

  // ---- phase 6: out_h = attn_h @ v_h  -> bP (pos is dead) ----------------
  {
    v8f d0 = {}, d1 = {};
    for (int t = 0; t < 8; ++t) {
      v2f a = ld2(sc + (h * 32 + 16 * mt + j) * SCS + 4 * t + 2 * p);
      int krow = 4 * t + 2 * p;
      v2f b0, b1;
      b0.x = bV[krow * STp + hb + j];
      b0.y = bV[(krow + 1) * STp + hb + j];
      b1.x = bV[krow * STp + hb + 16 + j];
      b1.y = bV[(krow + 1) * STp + hb + 16 + j];
      d0 = wmma4(a, b0, d0);
      d1 = wmma4(a, b1, d1);
    }
#pragma unroll
    for (int r = 0; r < 8; ++r) {
      int row = (16 * mt + r + 8 * p) * STp + hb;
      bP[row + j]      = d0[r];
      bP[row + 16 + j] = d1[r];
    }
  }
  __syncthreads();

  // ---- phase 7: final = out[32,128] @ Wo, maxpool over 32 rows, store ----
  {
    v8f d0 = {}, d1 = {};
    for (int t = 0; t < 32; ++t) {
      v2f bb = ld2(WoS + ((t * 2 + p) * 128 + nb + j) * 2);
      v2f a0 = ld2(bP + j * STp + 4 * t + 2 * p);
      v2f a1 = ld2(bP + (16 + j) * STp + 4 * t + 2 * p);
      d0 = wmma4(a0, bb, d0);
      d1 = wmma4(a1, bb, d1);
    }
    float cmax = -3.0e38f;
#pragma unroll
    for (int r = 0; r < 8; ++r) cmax = fmaxf(cmax, fmaxf(d0[r], d1[r]));
    cmax = fmaxf(cmax, __shfl_xor(cmax, 16, 32));   // combine row halves
    if (p == 0)
      out_pts[((size_t)b * DM + nb + j) * SS + s] = cmax;
  }
}

// ---------------------------------------------------------------------------
// Host launcher. d_out = [new_xyz (16*3*1024)] ++ [new_points (16*128*1024)].
// Workspace: fps idx | ball idx | swizzled Wq/Wk/Wv/Wp1/Wp2/Wo (~2.3 MB).
// ---------------------------------------------------------------------------
extern "C" void kernel_launch(void* const* d_in, const int* in_sizes, int n_in,
                              void* d_out, int out_size, void* d_ws, size_t ws_size,
                              hipStream_t stream) {
  const float* xyz    = (const float*)d_in[0];
  const float* points = (const float*)d_in[1];
  const float* Wq     = (const float*)d_in[2];
  const float* Wk     = (const float*)d_in[3];
  const float* Wv     = (const float*)d_in[4];
  const float* Wo     = (const float*)d_in[5];
  const float* Wp1    = (const float*)d_in[6];
  const float* bp1    = (const float*)d_in[7];
  const float* Wp2    = (const float*)d_in[8];
  const float* bp2    = (const float*)d_in[9];
  float* out = (float*)d_out;

  char* ws = (char*)d_ws;
  int*   fps  = (int*)(ws + 0);                 //  16*1024 ints      (64 KB)
  int*   ball = (int*)(ws + (64 << 10));        //  16*1024*32 ints   (2 MB)
  float* WqS  = (float*)(ws + 2162688);         //  8704 f32 (swizzled, K-pad 68)
  float* WkS  = (float*)(ws + 2197504);         //  8704 f32
  float* WvS  = (float*)(ws + 2232320);         //  8704 f32
  float* Wp1S = (float*)(ws + 2267136);         //   512 f32 (bias row folded)
  float* Wp2S = (float*)(ws + 2269184);         // 16384 f32
  float* WoS  = (float*)(ws + 2334720);         // 16384 f32  -> end ~2.29 MB

  prep_kernel<<<232, 256, 0, stream>>>(Wq, Wk, Wv, Wp1, bp1, Wp2, Wo,
                                       WqS, WkS, WvS, Wp1S, Wp2S, WoS);
  fps_kernel<<<NB, 1024, 0, stream>>>(xyz, fps);
  ballq_kernel<<<(NB * SS) / 8, 256, 0, stream>>>(xyz, fps, ball, out);

  const size_t smem =
      (size_t)(4 * 32 * STp + 4 * 32 * SCS + 32 * CINP + 128 + 32) * sizeof(float);
  tsa_main<<<NB * SS, 256, smem, stream>>>(xyz, points, WqS, WkS, WvS, WoS, Wp1S,
                                           Wp2S, bp2, fps, ball, out + NB * 3 * SS);
}